// SAGE_Conv_layer_34213709480236
// MI455X (gfx1250) — compile-verified
//
#include <hip/hip_runtime.h>
#include <hip/hip_bf16.h>

typedef __attribute__((ext_vector_type(2))) float v2f;
typedef __attribute__((ext_vector_type(8))) float v8f;

#define DFEAT 512
#define KCHUNK 256
#define LDS_STRIDE 260   // 256 + 4 pad: banks (4m+c)%64 distinct per half, float4-aligned

// ---------------- zero workspace ----------------
__global__ void sage_zero_f32(float* __restrict__ p, size_t n) {
    size_t i = (size_t)blockIdx.x * blockDim.x + threadIdx.x;
    size_t stride = (size_t)gridDim.x * blockDim.x;
    for (; i < n; i += stride) p[i] = 0.0f;
}

// ---------------- pack W_self/W_neigh into WMMA-native B layout ----------------
// wpk[n][p] = float4{ Wself[2p][n], Wself[2p+1][n], Wneigh[2p][n], Wneigh[2p+1][n] }
// so one b128 per lane per K-step feeds BOTH wmma ops.
__global__ __launch_bounds__(256) void sage_packW(
    const float* __restrict__ Wself, const float* __restrict__ Wneigh,
    float4* __restrict__ wpk)
{
    const int n = blockIdx.x;      // output column, 0..511
    const int p = threadIdx.x;     // K-pair, 0..255
    float4 v;
    v.x = Wself [(size_t)(2 * p    ) * DFEAT + n];
    v.y = Wself [(size_t)(2 * p + 1) * DFEAT + n];
    v.z = Wneigh[(size_t)(2 * p    ) * DFEAT + n];
    v.w = Wneigh[(size_t)(2 * p + 1) * DFEAT + n];
    wpk[(size_t)n * (DFEAT / 2) + p] = v;
}

// ---------------- edge scatter: msg_sum[dst] += x[src], deg[dst] += 1 ----------------
__global__ __launch_bounds__(128) void sage_scatter(
    const float* __restrict__ x, const int* __restrict__ src, const int* __restrict__ dst,
    float* __restrict__ msum, float* __restrict__ deg, int n_edges)
{
    int e = blockIdx.x;
    if (e >= n_edges) return;
    int s = src[e];
    int d = dst[e];
    const float4 v = ((const float4*)(x + (size_t)s * DFEAT))[threadIdx.x];
    float* drow = msum + (size_t)d * DFEAT + threadIdx.x * 4;
    atomicAdd(drow + 0, v.x);
    atomicAdd(drow + 1, v.y);
    atomicAdd(drow + 2, v.z);
    atomicAdd(drow + 3, v.w);
    if (threadIdx.x == 0) atomicAdd(deg + d, 1.0f);
}

// ---------------- fused GEMM + bias + leaky_relu + row L2-normalize ----------------
// block = 256 threads = 8 waves; block computes 16 full rows x 512 cols.
// wave w owns column tiles [4w, 4w+4); fp32 WMMA 16x16x4, K staged in LDS chunks of 256.
__global__ __launch_bounds__(256) void sage_gemm(
    const float* __restrict__ x, const float* __restrict__ msum, const float* __restrict__ deg,
    const float4* __restrict__ wpk, const float* __restrict__ bias, float* __restrict__ out)
{
    __shared__ float xs[16 * LDS_STRIDE];   // x rows, K-chunk
    __shared__ float ns[16 * LDS_STRIDE];   // mean-aggregated neighbor rows, K-chunk
    __shared__ float rowsq[16];             // per-row sum of squares

    const int tid  = threadIdx.x;
    const int wave = tid >> 5;
    const int lane = tid & 31;
    const int half = lane >> 4;   // 0: K+0/1 & rows 0..7 ; 1: K+2/3 & rows 8..15
    const int l16  = lane & 15;
    const int row0 = blockIdx.x * 16;

    if (tid < 16) rowsq[tid] = 0.0f;

    v8f acc[4];
    #pragma unroll
    for (int t = 0; t < 4; ++t) acc[t] = (v8f){0.f,0.f,0.f,0.f,0.f,0.f,0.f,0.f};

    for (int kb = 0; kb < DFEAT; kb += KCHUNK) {
        // ---- stage 16 x 256 slices of x and neigh (mean applied here) ----
        #pragma unroll
        for (int i = 0; i < 4; ++i) {
            int idx4 = tid + (i << 8);          // 0..1023 float4 slots
            int r  = idx4 >> 6;                 // 64 float4 per row
            int c4 = idx4 & 63;
            const size_t gbase = (size_t)(row0 + r) * DFEAT + kb;
            float4 xv = ((const float4*)(x    + gbase))[c4];
            float4 mv = ((const float4*)(msum + gbase))[c4];
            float invd = 1.0f / fmaxf(deg[row0 + r], 1.0f);
            ((float4*)(xs + r * LDS_STRIDE))[c4] = xv;
            float4 nv; nv.x = mv.x*invd; nv.y = mv.y*invd; nv.z = mv.z*invd; nv.w = mv.w*invd;
            ((float4*)(ns + r * LDS_STRIDE))[c4] = nv;
        }
        __syncthreads();

        // ---- accumulate: acc[t] += xA*Wself_tile + nA*Wneigh_tile over this K-chunk ----
        #pragma unroll
        for (int t = 0; t < 4; ++t) {
            const int n0 = (wave * 4 + t) * 16;
            // lane's packed-B stream: column n0+l16, starting K-pair (kb/2 + half)
            const float4* bp = wpk + (size_t)(n0 + l16) * (DFEAT / 2) + (kb >> 1) + half;
            v8f c = acc[t];
            for (int k = 0; k < KCHUNK; k += 4) {
                const int ka = k + 2 * half;              // this lane's K offset in chunk
                v2f ax = *(const v2f*)(xs + l16 * LDS_STRIDE + ka);
                v2f an = *(const v2f*)(ns + l16 * LDS_STRIDE + ka);
                const float4 bv = bp[k >> 1];             // one b128: both B fragments
                v2f bs; bs.x = bv.x; bs.y = bv.y;
                v2f bn; bn.x = bv.z; bn.y = bv.w;
                c = __builtin_amdgcn_wmma_f32_16x16x4_f32(false, ax, false, bs,
                                                          (short)0, c, false, false);
                c = __builtin_amdgcn_wmma_f32_16x16x4_f32(false, an, false, bn,
                                                          (short)0, c, false, false);
            }
            acc[t] = c;
        }
        __syncthreads();   // protect LDS before next chunk's restage
    }

    // ---- epilogue: bias + leaky_relu, per-row sum of squares ----
    float sq[8];
    #pragma unroll
    for (int r = 0; r < 8; ++r) sq[r] = 0.0f;

    #pragma unroll
    for (int t = 0; t < 4; ++t) {
        const float bv = bias[(wave * 4 + t) * 16 + l16];  // lane's column
        #pragma unroll
        for (int r = 0; r < 8; ++r) {
            float v = acc[t][r] + bv;
            v = (v >= 0.0f) ? v : 0.01f * v;               // leaky_relu
            acc[t][r] = v;
            sq[r] += v * v;
        }
    }
    // reduce across the 16 lanes of each half (xor masks stay within half)
    #pragma unroll
    for (int r = 0; r < 8; ++r) {
        float s = sq[r];
        s += __shfl_xor(s, 1);
        s += __shfl_xor(s, 2);
        s += __shfl_xor(s, 4);
        s += __shfl_xor(s, 8);
        sq[r] = s;
    }
    if (l16 == 0) {
        #pragma unroll
        for (int r = 0; r < 8; ++r) atomicAdd(&rowsq[r + 8 * half], sq[r]);
    }
    __syncthreads();

    // ---- scale by 1/max(||row||, eps) and store ----
    float inv[8];
    #pragma unroll
    for (int r = 0; r < 8; ++r)
        inv[r] = 1.0f / fmaxf(sqrtf(rowsq[r + 8 * half]), 1e-12f);

    #pragma unroll
    for (int t = 0; t < 4; ++t) {
        const int n0 = (wave * 4 + t) * 16;
        #pragma unroll
        for (int r = 0; r < 8; ++r) {
            const int row = row0 + r + 8 * half;
            out[(size_t)row * DFEAT + n0 + l16] = acc[t][r] * inv[r];
        }
    }
}

extern "C" void kernel_launch(void* const* d_in, const int* in_sizes, int n_in,
                              void* d_out, int out_size, void* d_ws, size_t ws_size,
                              hipStream_t stream) {
    (void)n_in; (void)out_size; (void)ws_size;
    const float* x      = (const float*)d_in[0];
    const int*   src    = (const int*)  d_in[1];
    const int*   dst    = (const int*)  d_in[2];
    const float* Wself  = (const float*)d_in[3];
    const float* Wneigh = (const float*)d_in[4];
    const float* bias   = (const float*)d_in[5];
    float* out = (float*)d_out;

    const int n_nodes = in_sizes[0] / DFEAT;   // 10000
    const int n_edges = in_sizes[1];           // 160000

    float*  msum = (float*)d_ws;                                   // [n_nodes, 512]
    float*  deg  = msum + (size_t)n_nodes * DFEAT;                 // [n_nodes]
    float4* wpk  = (float4*)(deg + n_nodes);                       // [512][256] float4

    const size_t zcount = (size_t)n_nodes * DFEAT + (size_t)n_nodes;
    sage_zero_f32<<<1024, 256, 0, stream>>>(msum, zcount);
    sage_packW<<<DFEAT, DFEAT / 2, 0, stream>>>(Wself, Wneigh, wpk);
    sage_scatter<<<n_edges, 128, 0, stream>>>(x, src, dst, msum, deg, n_edges);
    sage_gemm<<<n_nodes / 16, 256, 0, stream>>>(x, msum, deg, wpk, bias, out);
}